// Model_20040317403656
// MI455X (gfx1250) — compile-verified
//
#include <hip/hip_runtime.h>

// ---------------------------------------------------------------------------
// Per-(B*C)-row 16-bin quantizer for x:(16,3,512,512) fp32.
// Memory-bound: ~100 MiB HBM traffic -> ~4.3us at 23.3 TB/s.
// Pass 1: row min/max (b128 loads, wave32 ds_swizzle reduce, ordered-uint atomics)
// Pass 2: quantize (double-buffered async global->LDS staging, NT stores)
// ---------------------------------------------------------------------------

typedef __attribute__((ext_vector_type(4))) float vfloat4;
typedef int v4i __attribute__((vector_size(16)));   // matches builtin param

#define ROWS        48
#define ROW_ELEMS   (512 * 512)          // 262144 floats per row
#define ROW_VEC     (ROW_ELEMS / 4)      // 65536 float4 per row
#define RED_BPR     64                   // reduce blocks per row
#define RED_VPB     (ROW_VEC / RED_BPR)  // 1024 float4 per reduce block
#define QAP_BPR     128                  // quantize blocks per row
#define QAP_VPB     (ROW_VEC / QAP_BPR)  // 512 float4 per quantize block (2 tiles)
#define FLT_EPS32   1.1920928955078125e-07f

// ---- order-preserving float <-> uint mapping (for integer min/max atomics) --
__device__ __forceinline__ unsigned ford(float f) {
  unsigned u = __float_as_uint(f);
  return (u & 0x80000000u) ? ~u : (u | 0x80000000u);
}
__device__ __forceinline__ float funord(unsigned o) {
  return __uint_as_float((o & 0x80000000u) ? (o ^ 0x80000000u) : ~o);
}

// ---- wave32 butterfly min/max via ds_swizzle (xor mode, and_mask=0x1f) -----
#define SWZ_MINMAX(MASK)                                                       \
  mn = fminf(mn, __int_as_float(__builtin_amdgcn_ds_swizzle(                   \
                   __float_as_int(mn), (((MASK) << 10) | 0x1f))));             \
  mx = fmaxf(mx, __int_as_float(__builtin_amdgcn_ds_swizzle(                   \
                   __float_as_int(mx), (((MASK) << 10) | 0x1f))));

// ---- CDNA5 async global->LDS copy (ASYNCcnt-tracked) -----------------------
__device__ __forceinline__ void async_copy_b128(const vfloat4* gsrc,
                                                vfloat4* lds_dst) {
#if __has_builtin(__builtin_amdgcn_global_load_async_to_lds_b128)
  __builtin_amdgcn_global_load_async_to_lds_b128(
      (__attribute__((address_space(1))) v4i*)gsrc,
      (__attribute__((address_space(3))) v4i*)lds_dst,
      0, 0);
#else
  unsigned lds_off =
      (unsigned)(size_t)(__attribute__((address_space(3))) void*)lds_dst;
  unsigned long long gaddr = (unsigned long long)gsrc;
  asm volatile("global_load_async_to_lds_b128 %0, %1, off"
               :: "v"(lds_off), "v"(gaddr)
               : "memory");
#endif
}

__device__ __forceinline__ void wait_async_le1() {
#if __has_builtin(__builtin_amdgcn_s_wait_asynccnt)
  __builtin_amdgcn_s_wait_asynccnt(1);
#else
  asm volatile("s_wait_asynccnt 1" ::: "memory");
#endif
  asm volatile("" ::: "memory");  // compiler-only fence: order the LDS reads
}
__device__ __forceinline__ void wait_async_le0() {
#if __has_builtin(__builtin_amdgcn_s_wait_asynccnt)
  __builtin_amdgcn_s_wait_asynccnt(0);
#else
  asm volatile("s_wait_asynccnt 0" ::: "memory");
#endif
  asm volatile("" ::: "memory");
}

// ---------------------------------------------------------------------------
// Kernel 0: init workspace (48 rows x {minOrd, maxOrd})
// ---------------------------------------------------------------------------
extern "C" __global__ void q_init(unsigned* __restrict__ ws) {
  int t = threadIdx.x;
  if (t < 2 * ROWS) ws[t] = (t & 1) ? 0u : 0xFFFFFFFFu;
}

// ---------------------------------------------------------------------------
// Kernel 1: per-row min/max reduction
// ---------------------------------------------------------------------------
extern "C" __global__ __launch_bounds__(256)
void q_minmax(const float* __restrict__ x, unsigned* __restrict__ ws) {
  const int row = blockIdx.x >> 6;     // / RED_BPR
  const int blk = blockIdx.x & 63;     // % RED_BPR
  const vfloat4* __restrict__ xr =
      (const vfloat4*)x + (size_t)row * ROW_VEC + (size_t)blk * RED_VPB;

  float mn =  3.402823466e38f;
  float mx = -3.402823466e38f;
#pragma unroll
  for (int i = 0; i < RED_VPB / 256; ++i) {
    vfloat4 v = xr[i * 256 + threadIdx.x];
    mn = fminf(mn, fminf(fminf(v.x, v.y), fminf(v.z, v.w)));
    mx = fmaxf(mx, fmaxf(fmaxf(v.x, v.y), fmaxf(v.z, v.w)));
  }

  // intra-wave (wave32) reduction
  SWZ_MINMAX(16) SWZ_MINMAX(8) SWZ_MINMAX(4) SWZ_MINMAX(2) SWZ_MINMAX(1)

  // cross-wave combine (8 waves / 256-thread block)
  __shared__ float smn[8], smx[8];
  const int lane = threadIdx.x & 31;
  const int wv   = threadIdx.x >> 5;
  if (lane == 0) { smn[wv] = mn; smx[wv] = mx; }
  __syncthreads();
  if (threadIdx.x == 0) {
#pragma unroll
    for (int w = 1; w < 8; ++w) {
      mn = fminf(mn, smn[w]);
      mx = fmaxf(mx, smx[w]);
    }
    atomicMin(ws + 2 * row + 0, ford(mn));
    atomicMax(ws + 2 * row + 1, ford(mx));
  }
}

// ---------------------------------------------------------------------------
// Kernel 2: quantize with double-buffered async global->LDS staging
// ---------------------------------------------------------------------------
__device__ __forceinline__ float quant1(float v, float mn, float inv,
                                        float width, bool degen) {
  float t = (v - mn) * inv;
  int r = (int)t;                       // trunc; t >= ~0 in-range
  r = r < 0 ? 0 : (r > 15 ? 15 : r);    // clamp to [0, 15]
  float q = fmaf((float)r + 0.5f, width, mn);   // bin midpoint
  return degen ? mn : q;
}

extern "C" __global__ __launch_bounds__(256)
void q_apply(const float* __restrict__ x, float* __restrict__ out,
             const unsigned* __restrict__ ws) {
  __shared__ vfloat4 stage[2][256];

  const int row = blockIdx.x >> 7;     // / QAP_BPR
  const int blk = blockIdx.x & 127;    // % QAP_BPR
  const int tid = threadIdx.x;

  const float mn = funord(ws[2 * row + 0]);
  const float mx = funord(ws[2 * row + 1]);
  const bool  degen = fabsf(mn - mx) <= (4.0f * FLT_EPS32 + 1e-5f * fabsf(mx));
  const float inv   = degen ? 0.0f : 16.0f / (mx - mn);
  const float width = (mx - mn) * (1.0f / 16.0f);

  const size_t base = (size_t)row * ROW_VEC + (size_t)blk * QAP_VPB;
  const vfloat4* __restrict__ xr   = (const vfloat4*)x + base;
  vfloat4*       __restrict__ outr = (vfloat4*)out + base;

  // Issue both tiles up front (each lane stages its own 16B slot).
  async_copy_b128(xr + tid,       &stage[0][tid]);
  async_copy_b128(xr + 256 + tid, &stage[1][tid]);

  // Tile 0: async loads complete in order -> asynccnt<=1 means tile 0 landed.
  wait_async_le1();
  {
    vfloat4 v = stage[0][tid];
    vfloat4 q;
    q.x = quant1(v.x, mn, inv, width, degen);
    q.y = quant1(v.y, mn, inv, width, degen);
    q.z = quant1(v.z, mn, inv, width, degen);
    q.w = quant1(v.w, mn, inv, width, degen);
    __builtin_nontemporal_store(q, outr + tid);        // NT: never re-read
  }

  // Tile 1
  wait_async_le0();
  {
    vfloat4 v = stage[1][tid];
    vfloat4 q;
    q.x = quant1(v.x, mn, inv, width, degen);
    q.y = quant1(v.y, mn, inv, width, degen);
    q.z = quant1(v.z, mn, inv, width, degen);
    q.w = quant1(v.w, mn, inv, width, degen);
    __builtin_nontemporal_store(q, outr + 256 + tid);
  }
}

// ---------------------------------------------------------------------------
extern "C" void kernel_launch(void* const* d_in, const int* in_sizes, int n_in,
                              void* d_out, int out_size, void* d_ws,
                              size_t ws_size, hipStream_t stream) {
  const float* x   = (const float*)d_in[0];
  float*       out = (float*)d_out;
  unsigned*    ws  = (unsigned*)d_ws;   // 96 uints: {minOrd, maxOrd} x 48 rows

  q_init  <<<1, 128, 0, stream>>>(ws);
  q_minmax<<<ROWS * RED_BPR, 256, 0, stream>>>(x, ws);
  q_apply <<<ROWS * QAP_BPR, 256, 0, stream>>>(x, out, ws);
}